// SpikeCNN_48378511622376
// MI455X (gfx1250) — compile-verified
//
#include <hip/hip_runtime.h>

typedef __attribute__((ext_vector_type(2))) float v2f;
typedef __attribute__((ext_vector_type(8))) float v8f;

#define TSTEPS 8
#define NTHR 224   // 7 waves (wave32); each wave owns M-tiles {w, w+7} -> 14 tiles, M-pad 224 >= 196

__global__ __launch_bounds__(NTHR) void snn_fused(
    const float* __restrict__ x, const float* __restrict__ w1,
    const float* __restrict__ b1, const float* __restrict__ w2,
    const float* __restrict__ b2, const float* __restrict__ wf,
    const float* __restrict__ bf, float* __restrict__ out, int Bimg)
{
  __shared__ float    sx[784];      // input image
  __shared__ float    sw1[75];
  __shared__ float    sb1[3];
  __shared__ unsigned sp1[588];     // packed 4-bit pool counts x 8 steps
  __shared__ float    p1pad[1296];  // [4][18][18]; ch3 + halo stay zero
  __shared__ float    s2buf[1176];  // [6][196] spikes
  __shared__ float    p2buf[294];
  __shared__ float    fcp[160];
  __shared__ float    swf[2940];
  __shared__ float    sbf[10];

  const int tid  = threadIdx.x;
  const int img  = blockIdx.x;
  const int lane = tid & 31;
  const int wv   = tid >> 5;          // wave id 0..6
  const int hi   = (lane >> 4) & 1;   // lane half
  const int col  = lane & 15;

  for (int i = tid; i < 784;  i += NTHR) sx[i]  = x[img*784 + i];
  for (int i = tid; i < 75;   i += NTHR) sw1[i] = w1[i];
  if (tid < 3)  sb1[tid] = b1[tid];
  for (int i = tid; i < 2940; i += NTHR) swf[i] = wf[i];
  if (tid < 10) sbf[tid] = bf[tid];
  for (int i = tid; i < 1296; i += NTHR) p1pad[i] = 0.0f;
  __syncthreads();

  // ---- conv1 once + full 8-step spike train, pooled counts packed 4b/step
  for (int i = tid; i < 588; i += NTHR) {
    int oc = i/196, pp = i%196, py = pp/14, px = pp%14;
    float y[4];
    #pragma unroll
    for (int d = 0; d < 4; ++d) {
      int oy = 2*py + (d>>1), ox = 2*px + (d&1);
      float acc = sb1[oc];
      #pragma unroll
      for (int ky = 0; ky < 5; ++ky) {
        int iy = oy + ky - 2;
        if (iy < 0 || iy > 27) continue;
        #pragma unroll
        for (int kx = 0; kx < 5; ++kx) {
          int ix = ox + kx - 2;
          if (ix < 0 || ix > 27) continue;
          acc += sw1[oc*25 + ky*5 + kx] * sx[iy*28 + ix];
        }
      }
      y[d] = acc;
    }
    float m0=0.f, m1=0.f, m2=0.f, m3=0.f;
    unsigned pack = 0u;
    #pragma unroll
    for (int t = 0; t < TSTEPS; ++t) {
      unsigned c = 0u;
      m0 += y[0]; if (m0 >= 1.0f) { m0 -= 1.0f; c++; }
      m1 += y[1]; if (m1 >= 1.0f) { m1 -= 1.0f; c++; }
      m2 += y[2]; if (m2 >= 1.0f) { m2 -= 1.0f; c++; }
      m3 += y[3]; if (m3 >= 1.0f) { m3 -= 1.0f; c++; }
      pack |= c << (4*t);
    }
    sp1[i] = pack;
  }

  // ---- conv2 weights as B fragments (w2col[75x6], zero-padded to 76x16)
  float bfrag0[19], bfrag1[19];
  #pragma unroll
  for (int kt = 0; kt < 19; ++kt) {
    int kk = 4*kt + 2*hi;
    bfrag0[kt] = (col < 6 && kk     < 75) ? w2[col*75 + kk]     : 0.0f;
    bfrag1[kt] = (col < 6 && kk + 1 < 75) ? w2[col*75 + kk + 1] : 0.0f;
  }
  const float b2lane = (col < 6) ? b2[col] : 0.0f;

  // per-lane register-resident im2col offsets (K index -> p1pad offset)
  int offA0[19], offA1[19];
  #pragma unroll
  for (int kt = 0; kt < 19; ++kt) {
    int k0 = 4*kt + 2*hi;
    int ic0 = k0/25, r0 = k0%25;
    offA0[kt] = ic0*324 + (r0/5)*18 + (r0%5);
    int k1 = k0 + 1;
    int ic1 = k1/25, r1 = k1%25;
    offA1[kt] = ic1*324 + (r1/5)*18 + (r1%5);   // k==75 -> ic 3 (zero channel)
  }

  // A-row base addresses for this lane's two M-tiles (clamped into valid rows)
  int baseA0, baseA1;
  {
    int p0 = wv*16 + col;       if (p0 > 195) p0 = 195;
    baseA0 = (p0/14)*18 + (p0%14);
    int p1 = (wv + 7)*16 + col; if (p1 > 195) p1 = 195;
    baseA1 = (p1/14)*18 + (p1%14);
  }

  v8f c0 = (v8f){0,0,0,0,0,0,0,0};   // mem2, M-tile wv
  v8f c1 = (v8f){0,0,0,0,0,0,0,0};   // mem2, M-tile wv+7

  float mem3 = 0.0f, accv = 0.0f;    // lanes 0..9 own FC state
  __syncthreads();

  for (int t = 0; t < TSTEPS; ++t) {
    // unpack pooled spike counts -> padded fp32 tile (interior only)
    for (int i = tid; i < 588; i += NTHR) {
      int ic = i/196, pp = i%196, py = pp/14, px = pp%14;
      unsigned c = (sp1[i] >> (4*t)) & 0xFu;
      p1pad[ic*324 + (py+2)*18 + (px+2)] = (float)c * 0.25f;
    }
    __syncthreads();

    // conv2 as implicit GEMM: mem2[224x16] += im2col(p1)[224x76] * w2col[76x16]
    // Two independent accumulator chains per wave, A loads double-buffered.
    {
      v2f a0, a1;
      a0.x = p1pad[baseA0 + offA0[0]]; a0.y = p1pad[baseA0 + offA1[0]];
      a1.x = p1pad[baseA1 + offA0[0]]; a1.y = p1pad[baseA1 + offA1[0]];
      #pragma unroll
      for (int kt = 0; kt < 19; ++kt) {
        v2f n0, n1;
        if (kt < 18) {
          n0.x = p1pad[baseA0 + offA0[kt+1]]; n0.y = p1pad[baseA0 + offA1[kt+1]];
          n1.x = p1pad[baseA1 + offA0[kt+1]]; n1.y = p1pad[baseA1 + offA1[kt+1]];
        } else { n0 = a0; n1 = a1; }
        v2f bb; bb.x = bfrag0[kt]; bb.y = bfrag1[kt];
        c0 = __builtin_amdgcn_wmma_f32_16x16x4_f32(false, a0, false, bb, (short)0, c0, false, false);
        c1 = __builtin_amdgcn_wmma_f32_16x16x4_f32(false, a1, false, bb, (short)0, c1, false, false);
        a0 = n0; a1 = n1;
      }
    }

    // + bias, fire threshold in-register, emit spikes to LDS
    #pragma unroll
    for (int r = 0; r < 8; ++r) {
      float v0 = c0[r] + b2lane;
      float s0 = (v0 >= 1.0f) ? 1.0f : 0.0f;
      c0[r] = v0 - s0;
      int pos0 = wv*16 + r + 8*hi;
      if (col < 6 && pos0 < 196) s2buf[col*196 + pos0] = s0;

      float v1 = c1[r] + b2lane;
      float s1 = (v1 >= 1.0f) ? 1.0f : 0.0f;
      c1[r] = v1 - s1;
      int pos1 = (wv + 7)*16 + r + 8*hi;
      if (col < 6 && pos1 < 196) s2buf[col*196 + pos1] = s1;
    }
    __syncthreads();

    // avgpool 2x2 -> p2[294] (flatten order oc,qy,qx matches reference)
    for (int i = tid; i < 294; i += NTHR) {
      int oc = i/49, pq = i%49, qy = pq/7, qx = pq%7;
      int b0 = oc*196 + (2*qy)*14 + 2*qx;
      p2buf[i] = 0.25f*(s2buf[b0] + s2buf[b0+1] + s2buf[b0+14] + s2buf[b0+15]);
    }
    __syncthreads();

    // FC partials: 10 outputs x 16 strided slices
    if (tid < 160) {
      int oc = tid >> 4, s = tid & 15;
      float part = 0.0f;
      for (int j = s; j < 294; j += 16) part += p2buf[j] * swf[oc*294 + j];
      fcp[tid] = part;
    }
    __syncthreads();
    if (tid < 10) {
      float sum = sbf[tid];
      #pragma unroll
      for (int k = 0; k < 16; ++k) sum += fcp[tid*16 + k];
      mem3 += sum;
      float sp = (mem3 >= 1.0f) ? 1.0f : 0.0f;
      mem3 -= sp; accv += sp;
      if (t == 3) out[img*10 + tid]           = accv * 0.25f;
      if (t == 7) out[Bimg*10 + img*10 + tid] = accv * 0.125f;
    }
    __syncthreads();
  }
}

extern "C" void kernel_launch(void* const* d_in, const int* in_sizes, int n_in,
                              void* d_out, int out_size, void* d_ws, size_t ws_size,
                              hipStream_t stream) {
  (void)n_in; (void)out_size; (void)d_ws; (void)ws_size;
  const float* x  = (const float*)d_in[0];
  const float* w1 = (const float*)d_in[1];
  const float* b1 = (const float*)d_in[2];
  const float* w2 = (const float*)d_in[3];
  const float* b2 = (const float*)d_in[4];
  const float* wf = (const float*)d_in[5];
  const float* bf = (const float*)d_in[6];
  float* out = (float*)d_out;
  int Bimg = in_sizes[0] / 784;          // B images of 1x28x28
  snn_fused<<<Bimg, NTHR, 0, stream>>>(x, w1, b1, w2, b2, wf, bf, out, Bimg);
}